// GraphNN_75857712382439
// MI455X (gfx1250) — compile-verified
//
#include <hip/hip_runtime.h>

// ---------------------------------------------------------------------------
// GraphNN on MI455X (gfx1250)
//
// logits[i,j] = s[j] is row-independent, so softmax can use the GLOBAL shift
// M = max_j s[j]; w[j] = exp(s[j]-M). Then
//   agg[i,:] = (A @ (w*vals))[i,:] / (A @ w)[i],  A = 0/1 adjacency,
// i.e. a binary-matrix GEMM done with V_WMMA_F32_16X16X4_F32 (exact f32).
// The j-sweep is split into NSEG segments for occupancy; partials are summed
// in a final kernel that also runs the decoder MLP.
// ---------------------------------------------------------------------------

#define N_NODES 8192
#define DIN     7
#define DH      8
#define DOUT    23
#define CUTOFF  3.6f
#define SLOPE   0.01f

#define CJ              512                      // j-chunk staged in LDS
#define NSEG            8                        // j-sweep segments
#define SEGJ            (N_NODES / NSEG)         // 1024
#define WAVES_PER_BLOCK 4
#define BLOCK_THREADS   (WAVES_PER_BLOCK * 32)
#define ROWS_PER_BLOCK  (WAVES_PER_BLOCK * 16)   // 64

typedef __attribute__((ext_vector_type(2))) float v2f;
typedef __attribute__((ext_vector_type(8))) float v8f;

__device__ __forceinline__ float leaky(float v) { return v >= 0.0f ? v : SLOPE * v; }

// ---------------------------------------------------------------------------
// Kernel 1: per-node MLP. arr1[j*8 + 0..6] = vals, arr1[j*8+7] = s = keys.qrys
// ---------------------------------------------------------------------------
__global__ __launch_bounds__(256) void node_mlp_kernel(
    const float* __restrict__ x,
    const float* __restrict__ W1, const float* __restrict__ b1,
    const float* __restrict__ W2, const float* __restrict__ b2,
    const float* __restrict__ W3, const float* __restrict__ b3,
    float* __restrict__ arr1)
{
    const int j = blockIdx.x * 256 + threadIdx.x;
    if (j >= N_NODES) return;

    float xv[DIN];
#pragma unroll
    for (int t = 0; t < DIN; ++t) xv[t] = x[j * DIN + t];

    float h1[DH];
#pragma unroll
    for (int k = 0; k < DH; ++k) {
        float a = b1[k];
#pragma unroll
        for (int t = 0; t < DIN; ++t) a += W1[k * DIN + t] * xv[t];
        h1[k] = leaky(a);
    }

    float h2[DH];
#pragma unroll
    for (int k = 0; k < DH; ++k) {
        float a = b2[k];
#pragma unroll
        for (int t = 0; t < DH; ++t) a += W2[k * DH + t] * h1[t];
        h2[k] = leaky(a);
    }

    float h3[DOUT];
#pragma unroll
    for (int k = 0; k < DOUT; ++k) {
        float a = b3[k];
#pragma unroll
        for (int t = 0; t < DH; ++t) a += W3[k * DH + t] * h2[t];
        h3[k] = a;                       // layer 3 has no activation
    }

    float s = 0.0f;
#pragma unroll
    for (int t = 0; t < 8; ++t) s += h3[DIN + t] * h3[DIN + 8 + t];

#pragma unroll
    for (int n = 0; n < 7; ++n) arr1[j * 8 + n] = h3[n];  // vals
    arr1[j * 8 + 7] = s;
}

// ---------------------------------------------------------------------------
// Kernel 2: global max of s (single block)
// ---------------------------------------------------------------------------
__global__ __launch_bounds__(256) void smax_kernel(const float* __restrict__ arr1,
                                                   float* __restrict__ Mptr)
{
    __shared__ float red[8];
    float m = -3.4e38f;
    for (int idx = threadIdx.x; idx < N_NODES; idx += 256)
        m = fmaxf(m, arr1[idx * 8 + 7]);
#pragma unroll
    for (int off = 16; off > 0; off >>= 1)
        m = fmaxf(m, __shfl_xor(m, off, 32));
    if ((threadIdx.x & 31) == 0) red[threadIdx.x >> 5] = m;
    __syncthreads();
    if (threadIdx.x == 0) {
        float mm = red[0];
#pragma unroll
        for (int w = 1; w < 8; ++w) mm = fmaxf(mm, red[w]);
        *Mptr = mm;
    }
}

// ---------------------------------------------------------------------------
// Kernel 3: in-place arr1[j] := [w*vals(7), w],  w = exp(s - M).
// Also packs coords into a float4 array for coalesced LDS staging.
// ---------------------------------------------------------------------------
__global__ __launch_bounds__(256) void bvec_kernel(float* __restrict__ arr1,
                                                   const float* __restrict__ Mptr,
                                                   const float* __restrict__ x,
                                                   float4* __restrict__ xc4)
{
    const int j = blockIdx.x * 256 + threadIdx.x;
    if (j >= N_NODES) return;
    const float M = *Mptr;
    const float w = __expf(arr1[j * 8 + 7] - M);
#pragma unroll
    for (int n = 0; n < 7; ++n) arr1[j * 8 + n] *= w;
    arr1[j * 8 + 7] = w;

    float4 c;
    c.x = x[j * DIN + 0];
    c.y = x[j * DIN + 1];
    c.z = x[j * DIN + 2];
    c.w = 0.0f;
    xc4[j] = c;
}

// ---------------------------------------------------------------------------
// Inner sweep over one LDS-staged chunk. DIAG enables the j!=i exclusion,
// needed only for the single chunk overlapping this block's 64 rows.
//
// A-fragment (ISA 16x4 f32): lanes 0-15 rows M=lane, K=0(v0),1(v1);
//                            lanes 16-31 rows M=lane-16, K=2(v0),3(v1).
// B-fragment (4x16): lane holds (K=2*half+{0,1}, N=n) in v0/v1; ldsB16 is
// padded to 16 cols (8..15 = 0) so loads are unconditional (no EXEC churn).
// ---------------------------------------------------------------------------
template <bool DIAG>
__device__ __forceinline__ void sweep_chunk(
    const float* ldsB16, const float4* ldsC, int cb, int i,
    float xi0, float xi1, float xi2, int half, int n, v8f& acc)
{
    for (int jj = 0; jj < CJ; jj += 4) {
        const int jl = jj + 2 * half;

        const float4 ca = ldsC[jl];
        const float4 cc = ldsC[jl + 1];

        const float d0 = fabsf(xi0 - ca.x) + fabsf(xi1 - ca.y) + fabsf(xi2 - ca.z);
        const float d1 = fabsf(xi0 - cc.x) + fabsf(xi1 - cc.y) + fabsf(xi2 - cc.z);

        bool m0 = d0 <= CUTOFF;
        bool m1 = d1 <= CUTOFF;
        if (DIAG) {
            const int j0 = cb + jl;
            m0 = m0 && (j0 != i);
            m1 = m1 && (j0 + 1 != i);
        }

        v2f a;
        a.x = m0 ? 1.0f : 0.0f;
        a.y = m1 ? 1.0f : 0.0f;

        v2f b;                                    // one ds_load_2addr_b32
        b.x = ldsB16[jl * 16 + n];
        b.y = ldsB16[jl * 16 + 16 + n];

        acc = __builtin_amdgcn_wmma_f32_16x16x4_f32(
            false, a, false, b, (short)0, acc, false, false);
    }
}

// ---------------------------------------------------------------------------
// Kernel 4: attention GEMM. Block = (row-group rg, j-segment seg).
// Writes per-row partial [num(7) | denom] for its segment.
// ---------------------------------------------------------------------------
__global__ __launch_bounds__(BLOCK_THREADS) void attn_kernel(
    const float* __restrict__ x,
    const float* __restrict__ bvec,     // N x 8: [w*vals | w]
    const float4* __restrict__ xc4,     // N packed coords
    float* __restrict__ partial)        // NSEG x N x 8
{
    __shared__ __align__(16) float  ldsB16[CJ * 16];   // 32 KB (cols 8..15 = 0)
    __shared__ __align__(16) float4 ldsC[CJ];          //  8 KB coords

    const int tid  = threadIdx.x;
    const int lane = tid & 31;
    const int wave = tid >> 5;
    const int half = lane >> 4;
    const int n    = lane & 15;

    const int rg      = blockIdx.x / NSEG;
    const int seg     = blockIdx.x % NSEG;
    const int rowBase = rg * ROWS_PER_BLOCK;
    const int i       = rowBase + wave * 16 + n;       // this lane's A row

    const float xi0 = x[i * DIN + 0];
    const float xi1 = x[i * DIN + 1];
    const float xi2 = x[i * DIN + 2];

    const int diagChunk = rowBase & ~(CJ - 1);         // chunk holding j==i

    v8f acc = {0.f, 0.f, 0.f, 0.f, 0.f, 0.f, 0.f, 0.f};

    const int jBeg = seg * SEGJ;
    const int jEnd = jBeg + SEGJ;
    for (int cb = jBeg; cb < jEnd; cb += CJ) {
        __syncthreads();
        // --- stage B chunk, padded to 16 floats/row ---
        {
            const float4* gB = (const float4*)(bvec + (size_t)cb * 8);
            float4* sB = (float4*)ldsB16;
            for (int t = tid; t < CJ * 4; t += BLOCK_THREADS) {
                const int j4 = t >> 2, q = t & 3;
                float4 v = {0.f, 0.f, 0.f, 0.f};
                if (q < 2) v = gB[j4 * 2 + q];
                sB[t] = v;
            }
        }
        // --- stage coords chunk (coalesced float4 copies) ---
        for (int t = tid; t < CJ; t += BLOCK_THREADS)
            ldsC[t] = xc4[cb + t];
        __syncthreads();

        // prefetch next B chunk (global_prefetch_b8)
        if (cb + CJ < jEnd)
            __builtin_prefetch((const void*)(bvec + (size_t)(cb + CJ) * 8), 0, 1);

        if (cb == diagChunk)
            sweep_chunk<true >(ldsB16, ldsC, cb, i, xi0, xi1, xi2, half, n, acc);
        else
            sweep_chunk<false>(ldsB16, ldsC, cb, i, xi0, xi1, xi2, half, n, acc);
    }

    // --- scatter C tile: vgpr r -> row (rowBase + wave*16 + r + 8*half), col n.
    // Cols 8..15 are the zero pad; only n<8 lanes store.
    if (n < 8) {
        float* prow = partial +
            ((size_t)seg * N_NODES + rowBase + wave * 16 + 8 * half) * 8 + n;
#pragma unroll
        for (int r = 0; r < 8; ++r)
            prow[(size_t)r * 8] = acc[r];
    }
}

// ---------------------------------------------------------------------------
// Kernel 5: sum segment partials, normalize, decoder MLP.
// ---------------------------------------------------------------------------
__global__ __launch_bounds__(256) void finish_kernel(
    const float* __restrict__ x, const float* __restrict__ partial,
    const float* __restrict__ We, const float* __restrict__ be,
    const float* __restrict__ Wd, const float* __restrict__ bd,
    float* __restrict__ out)
{
    const int i = blockIdx.x * 256 + threadIdx.x;
    if (i >= N_NODES) return;

    float num[8] = {0.f, 0.f, 0.f, 0.f, 0.f, 0.f, 0.f, 0.f};
#pragma unroll
    for (int seg = 0; seg < NSEG; ++seg) {
        const float4* p = (const float4*)(partial + ((size_t)seg * N_NODES + i) * 8);
        const float4 a = p[0];
        const float4 b = p[1];
        num[0] += a.x; num[1] += a.y; num[2] += a.z; num[3] += a.w;
        num[4] += b.x; num[5] += b.y; num[6] += b.z; num[7] += b.w;
    }
    const float denom = fmaxf(num[7], 1e-30f);

    float inp[2 * DIN];
#pragma unroll
    for (int t = 0; t < DIN; ++t) inp[t] = x[i * DIN + t];
#pragma unroll
    for (int t = 0; t < DIN; ++t) inp[DIN + t] = num[t] / denom;

    float codes[DH];
#pragma unroll
    for (int k = 0; k < DH; ++k) {
        float a = be[k];
#pragma unroll
        for (int t = 0; t < 2 * DIN; ++t) a += We[k * (2 * DIN) + t] * inp[t];
        codes[k] = leaky(a);
    }
#pragma unroll
    for (int m = 0; m < DIN; ++m) {
        float a = bd[m];
#pragma unroll
        for (int k = 0; k < DH; ++k) a += Wd[m * DH + k] * codes[k];
        out[i * DIN + m] = a;
    }
}

// ---------------------------------------------------------------------------
extern "C" void kernel_launch(void* const* d_in, const int* in_sizes, int n_in,
                              void* d_out, int out_size, void* d_ws, size_t ws_size,
                              hipStream_t stream)
{
    (void)in_sizes; (void)n_in; (void)out_size; (void)ws_size;

    const float* x  = (const float*)d_in[0];
    const float* W1 = (const float*)d_in[1];
    const float* b1 = (const float*)d_in[2];
    const float* W2 = (const float*)d_in[3];
    const float* b2 = (const float*)d_in[4];
    const float* W3 = (const float*)d_in[5];
    const float* b3 = (const float*)d_in[6];
    const float* We = (const float*)d_in[7];
    const float* be = (const float*)d_in[8];
    const float* Wd = (const float*)d_in[9];
    const float* bd = (const float*)d_in[10];
    float* out = (float*)d_out;

    float*  arr1    = (float*)d_ws;                          // N*8
    float*  Mptr    = arr1 + (size_t)N_NODES * 8;            // 1 (16-float slot)
    float4* xc4     = (float4*)(arr1 + (size_t)N_NODES * 8 + 16);   // N float4
    float*  partial = (float*)(xc4 + N_NODES);               // NSEG*N*8

    node_mlp_kernel<<<N_NODES / 256, 256, 0, stream>>>(x, W1, b1, W2, b2, W3, b3, arr1);
    smax_kernel<<<1, 256, 0, stream>>>(arr1, Mptr);
    bvec_kernel<<<N_NODES / 256, 256, 0, stream>>>(arr1, Mptr, x, xc4);
    attn_kernel<<<(N_NODES / ROWS_PER_BLOCK) * NSEG, BLOCK_THREADS, 0, stream>>>(
        x, arr1, xc4, partial);
    finish_kernel<<<N_NODES / 256, 256, 0, stream>>>(x, partial, We, be, Wd, bd, out);
}